// CDAutoEncoder_32667521254225
// MI455X (gfx1250) — compile-verified
//
#include <hip/hip_runtime.h>

typedef __attribute__((ext_vector_type(2))) float v2f;
typedef __attribute__((ext_vector_type(4))) float v4f;
typedef __attribute__((ext_vector_type(8))) float v8f;

#define B_    32
#define CIN_  16
#define H_    256
#define COUT_ 32
#define OH_   128
#define OW_   128

// LDS layout for the masked input row-pair: [cin][kh][col], padded so that the
// kh=1 half (used by lanes 16-31 of each wave) lands on a disjoint bank set.
// 288 floats stride: 288 % 64 == 32 banks shift -> conflict-free ds_load_b64.
#define KH_STRIDE  288
#define CIN_STRIDE (2 * KH_STRIDE)   // 576 floats per cin

__global__ __launch_bounds__(256) void conv2x2_relu_wmma(
    const float* __restrict__ x,     // [32,16,256,256]
    const float* __restrict__ noise, // [32,16,256,256]
    const float* __restrict__ W,     // [32,16,2,2] -> flat [co][k], k = cin*4+kh*2+kw
    const float* __restrict__ bias,  // [32]
    float* __restrict__ out)         // [32,32,128,128]
{
  __shared__ alignas(16) float lds[CIN_ * CIN_STRIDE]; // 36 KB

  const int blk = blockIdx.x;       // b*OH + oh
  const int b   = blk >> 7;
  const int oh  = blk & 127;
  const int t   = threadIdx.x;

  // ---------- Phase 1: cooperative masked load of the input row-pair ----------
  // 16 cin * 2 rows * 256 cols = 8192 floats; 256 threads * 8 iters * float4.
  {
    const size_t xbase = (size_t)b * CIN_ * (size_t)H_ * H_ + (size_t)(2 * oh) * H_;
#pragma unroll
    for (int i = 0; i < 8; ++i) {
      int idx = i * 1024 + t * 4;        // flat [cin][r][col], col multiple of 4
      int cin = idx >> 9;
      int rem = idx & 511;
      int r   = rem >> 8;
      int col = rem & 255;
      size_t g = xbase + (size_t)cin * ((size_t)H_ * H_) + (size_t)r * H_ + col;
      v4f xv = *(const v4f*)(x + g);
      v4f nv = *(const v4f*)(noise + g);
      v4f mv;
      mv.x = (nv.x > -0.1f) ? xv.x : 0.0f;
      mv.y = (nv.y > -0.1f) ? xv.y : 0.0f;
      mv.z = (nv.z > -0.1f) ? xv.z : 0.0f;
      mv.w = (nv.w > -0.1f) ? xv.w : 0.0f;
      *(v4f*)(&lds[cin * CIN_STRIDE + r * KH_STRIDE + col]) = mv;
    }
  }
  __syncthreads();

  // ---------- Phase 2: per-wave 32(co) x 16(ow) GEMM tile via f32 WMMA ----------
  const int lane = t & 31;
  const int wv   = t >> 5;        // 8 waves -> 8 x 16 = 128 = OW
  const int n    = lane & 15;     // N index (ow within tile) / M index for A frag
  const int hi   = lane >> 4;     // selects K pair {0,1} vs {2,3} in A/B fragments
  const int ow0  = wv * 16;

  // A fragments: ISA 16x4 f32 A layout -> lane(m=n, hi) holds K = 2*hi, 2*hi+1.
  // k-step kk corresponds to cin = kk; element pair is 8B-aligned -> b64 loads.
  v2f a0[16], a1[16];
#pragma unroll
  for (int kk = 0; kk < 16; ++kk) {
    int kb = kk * 4 + hi * 2;
    a0[kk] = *(const v2f*)(W + (n       * 64 + kb));
    a1[kk] = *(const v2f*)(W + ((n + 16) * 64 + kb));
  }

  v8f c0 = {};
  v8f c1 = {};
  const float* bbase = &lds[hi * KH_STRIDE + 2 * (ow0 + n)];
#pragma unroll
  for (int kk = 0; kk < 16; ++kk) {
    // B 4x16 fragment: lanes 0-15 rows {0,1}, lanes 16-31 rows {2,3};
    // rows (kh,kw) pairs are adjacent in LDS -> single ds_load_b64 per lane.
    v2f bf = *(const v2f*)(bbase + kk * CIN_STRIDE);
    c0 = __builtin_amdgcn_wmma_f32_16x16x4_f32(
        false, a0[kk], false, bf, (short)0, c0, false, false);
    c1 = __builtin_amdgcn_wmma_f32_16x16x4_f32(
        false, a1[kk], false, bf, (short)0, c1, false, false);
  }

  // ---------- Phase 3: bias + ReLU + store ----------
  // C/D f32 16x16 layout: VGPR r -> (M=r, N=lane) for lanes 0-15,
  //                                  (M=r+8, N=lane-16) for lanes 16-31.
#pragma unroll
  for (int r = 0; r < 8; ++r) {
    int co0 = hi * 8 + r;
    float v0 = c0[r] + bias[co0];
    v0 = v0 > 0.0f ? v0 : 0.0f;
    out[(((size_t)b * COUT_ + co0) * OH_ + oh) * OW_ + ow0 + n] = v0;

    int co1 = co0 + 16;
    float v1 = c1[r] + bias[co1];
    v1 = v1 > 0.0f ? v1 : 0.0f;
    out[(((size_t)b * COUT_ + co1) * OH_ + oh) * OW_ + ow0 + n] = v1;
  }
}

extern "C" void kernel_launch(void* const* d_in, const int* in_sizes, int n_in,
                              void* d_out, int out_size, void* d_ws, size_t ws_size,
                              hipStream_t stream) {
  const float* x     = (const float*)d_in[0];
  const float* noise = (const float*)d_in[1];
  const float* W     = (const float*)d_in[2];
  const float* bias  = (const float*)d_in[3];
  float* out         = (float*)d_out;

  dim3 grid(B_ * OH_);   // 4096 blocks: one per (batch, output row)
  dim3 block(256);       // 8 wave32 -> 8 x 16 = 128 output columns
  conv2x2_relu_wmma<<<grid, block, 0, stream>>>(x, noise, W, bias, out);
}